// Embedding_23338852286583
// MI455X (gfx1250) — compile-verified
//
#include <hip/hip_runtime.h>

// out[t, :] = b + W[VOCAB + age[t], :] + sum_j W[word[t, j], :]
// T = B*S tokens, D = 768, BAG = 40. One wave32 per token.
//
// Memory-bound gather: W (4.2 MB) is L2-resident; ~1 GB of gathered row
// traffic out of L2 plus a 25 MB streaming output write. Strategy:
//  - preload all 40 bag indices into 2 VGPRs (2 coalesced b32 loads),
//    extract with v_readlane -> no load/wait in the index path
//  - scalar (saddr) row base + 6x global_load_b128 per row, s_clause'd
//  - WGP-scope global_prefetch_b8, 4 rows ahead, lane-strided 96B so a
//    wave's 32 prefetches cover the whole 3 KB row

#define VOCAB      1276
#define AGE_DEPTH  91
#define DIM        768
#define BAG        40
#define WAVES_PER_BLOCK 8
#define CHUNKS     (DIM / 128)   // 6 float4-chunks per lane
#define PF         4             // prefetch distance (rows)

__device__ __forceinline__ void pf_row(const float* __restrict__ W, int idx, int lane) {
  const char* p = (const char*)(W + (size_t)idx * DIM);
  __builtin_prefetch(p + lane * 96, 0, 3);  // WGP scope: pull into near cache
}

__global__ __launch_bounds__(WAVES_PER_BLOCK * 32)
void emb_bag_gather_kernel(const int* __restrict__ word,   // [T, BAG]
                           const int* __restrict__ age,    // [T]
                           const float* __restrict__ W,    // [VOCAB+AGE_DEPTH, DIM]
                           const float* __restrict__ bias, // [DIM]
                           float* __restrict__ out,        // [T, DIM]
                           int T) {
  const int wave = blockIdx.x * WAVES_PER_BLOCK + (threadIdx.x >> 5);
  const int lane = threadIdx.x & 31;
  if (wave >= T) return;

  const int* wptr = word + (size_t)wave * BAG;

  // Preload all 40 indices in two coalesced loads:
  //   vlo: lane l  -> wptr[l]         (rows 0..31)
  //   vhi: lane l  -> wptr[32+(l&7)]  (rows 32..39, replicated 4x)
  int vlo = wptr[lane];
  int vhi = wptr[32 + (lane & 7)];
  int aval = age[wave];

  // Accumulators: 6 x float4 per lane cover one 768-float output row.
  float4 acc[CHUNKS];
  const float4* b4 = (const float4*)bias;
#pragma unroll
  for (int c = 0; c < CHUNKS; ++c) acc[c] = b4[c * 32 + lane];

  // Kick off prefetches for the first PF bag rows.
#pragma unroll
  for (int p = 0; p < PF; ++p)
    pf_row(W, __builtin_amdgcn_readlane(vlo, p), lane);

  // Age row (accumulating it first gives the prefetches time to land).
  {
    int arow = VOCAB + __builtin_amdgcn_readfirstlane(aval);
    const float4* r = (const float4*)(W + (size_t)arow * DIM);
#pragma unroll
    for (int c = 0; c < CHUNKS; ++c) {
      float4 v = r[c * 32 + lane];
      acc[c].x += v.x; acc[c].y += v.y; acc[c].z += v.z; acc[c].w += v.w;
    }
  }

  // Bag rows 0..31: index via v_readlane (no memory dep), PF rows ahead.
#pragma unroll 4
  for (int j = 0; j < 32; ++j) {
    int idx = __builtin_amdgcn_readlane(vlo, j);
    // (j+PF)&31 wraps onto already-hot rows at the tail -> harmless.
    pf_row(W, __builtin_amdgcn_readlane(vlo, (j + PF) & 31), lane);
    const float4* r = (const float4*)(W + (size_t)idx * DIM);
#pragma unroll
    for (int c = 0; c < CHUNKS; ++c) {
      float4 v = r[c * 32 + lane];
      acc[c].x += v.x; acc[c].y += v.y; acc[c].z += v.z; acc[c].w += v.w;
    }
  }

  // Prefetch the 8 tail rows, then accumulate them.
#pragma unroll
  for (int p = 0; p < 8; ++p)
    pf_row(W, __builtin_amdgcn_readlane(vhi, p), lane);

#pragma unroll 4
  for (int j = 0; j < 8; ++j) {
    int idx = __builtin_amdgcn_readlane(vhi, j);
    const float4* r = (const float4*)(W + (size_t)idx * DIM);
#pragma unroll
    for (int c = 0; c < CHUNKS; ++c) {
      float4 v = r[c * 32 + lane];
      acc[c].x += v.x; acc[c].y += v.y; acc[c].z += v.z; acc[c].w += v.w;
    }
  }

  float4* o = (float4*)(out + (size_t)wave * DIM);
#pragma unroll
  for (int c = 0; c < CHUNKS; ++c) o[c * 32 + lane] = acc[c];
}

extern "C" void kernel_launch(void* const* d_in, const int* in_sizes, int n_in,
                              void* d_out, int out_size, void* d_ws, size_t ws_size,
                              hipStream_t stream) {
  (void)n_in; (void)d_ws; (void)ws_size; (void)out_size;
  const int*   word = (const int*)d_in[0];   // [B*S*BAG] int32
  const int*   age  = (const int*)d_in[1];   // [B*S] int32
  const float* W    = (const float*)d_in[2]; // [(VOCAB+AGE_DEPTH)*DIM] f32
  const float* bias = (const float*)d_in[3]; // [DIM] f32
  float*       out  = (float*)d_out;         // [B*S*DIM] f32

  const int T = in_sizes[1];  // number of tokens (B*S)
  const int blocks = (T + WAVES_PER_BLOCK - 1) / WAVES_PER_BLOCK;
  emb_bag_gather_kernel<<<blocks, WAVES_PER_BLOCK * 32, 0, stream>>>(
      word, age, W, bias, out, T);
}